// Teacher_Model_GCL_73890617360939
// MI455X (gfx1250) — compile-verified
//
#include <hip/hip_runtime.h>
#include <math.h>

#define N_USERS 100000
#define N_ITEMS 50000
#define DIM 64
#define IMG_D 1024
#define TXT_D 384

typedef float v2f __attribute__((ext_vector_type(2)));
typedef float v8f __attribute__((ext_vector_type(8)));

// Guaranteed-native f32 atomic add (no CAS fallback). No-return form -> STOREcnt;
// kernel-end S_ENDPGM implicitly waits. DEV scope for cross-WGP coherence.
__device__ __forceinline__ void atomicAddF(float* p, float v) {
  asm volatile("global_atomic_add_f32 %0, %1, off scope:SCOPE_DEV"
               :: "v"(p), "v"(v) : "memory");
}

// ---------------- zero fill ----------------
__global__ void k_zero(float* __restrict__ p, long n) {
  long i = (long)blockIdx.x * blockDim.x + threadIdx.x;
  if (i < n) p[i] = 0.0f;
}

// ---------------- GEMM: C[M x 64] = A[M x K] @ W[K x 64] + bias ----------------
// fp32 WMMA 16x16x4. One wave per 16-row tile, 4 N-tiles of 16 cols (D=64).
// A frag layout (ISA 7.12.2): lane&15 = M row; lane>>4 selects K pair {0,1}/{2,3}.
__global__ void __launch_bounds__(256)
k_gemm_bias_wmma(const float* __restrict__ A, const float* __restrict__ W,
                 const float* __restrict__ bias, float* __restrict__ C,
                 int M, int K) {
  int wave = (int)((blockIdx.x * blockDim.x + threadIdx.x) >> 5);
  int lane = threadIdx.x & 31;
  int row0 = wave * 16;
  if (row0 + 16 > M) return;

  int m  = lane & 15;   // A: row within tile; B/C: col within N-tile
  int kq = lane >> 4;   // 0 -> K{0,1}, 1 -> K{2,3}

  v8f acc0 = {}, acc1 = {}, acc2 = {}, acc3 = {};

  const float* ap = A + (size_t)(row0 + m) * K + kq * 2;
#pragma unroll 4
  for (int k = 0; k < K; k += 4) {
    v2f a = *(const v2f*)(ap + k);  // contiguous K pair, 8B load per lane
    const float* wp = W + (size_t)(k + kq * 2) * DIM + m;
    v2f b0 = { wp[0],  wp[DIM + 0]  };
    v2f b1 = { wp[16], wp[DIM + 16] };
    v2f b2 = { wp[32], wp[DIM + 32] };
    v2f b3 = { wp[48], wp[DIM + 48] };
    acc0 = __builtin_amdgcn_wmma_f32_16x16x4_f32(false, a, false, b0, (short)0, acc0, false, false);
    acc1 = __builtin_amdgcn_wmma_f32_16x16x4_f32(false, a, false, b1, (short)0, acc1, false, false);
    acc2 = __builtin_amdgcn_wmma_f32_16x16x4_f32(false, a, false, b2, (short)0, acc2, false, false);
    acc3 = __builtin_amdgcn_wmma_f32_16x16x4_f32(false, a, false, b3, (short)0, acc3, false, false);
  }

  float bi0 = bias[m], bi1 = bias[16 + m], bi2 = bias[32 + m], bi3 = bias[48 + m];
  // C/D layout: VGPR i holds M=i (lanes 0-15) / M=8+i (lanes 16-31), N=lane&15
#pragma unroll
  for (int i = 0; i < 8; ++i) {
    int r = row0 + kq * 8 + i;
    float* cp = C + (size_t)r * DIM;
    cp[m]      = acc0[i] + bi0;
    cp[16 + m] = acc1[i] + bi1;
    cp[32 + m] = acc2[i] + bi2;
    cp[48 + m] = acc3[i] + bi3;
  }
}

// ---------------- SpMM: out[dst[e]] += val[e] * X[src[e]]  (rows of 64) ----------------
// One wave32 per edge; scalarized edge metadata (readfirstlane -> SMEM loads),
// float2/lane = coalesced 256B row gather; native f32 atomic scatter.
__global__ void __launch_bounds__(256)
k_spmm(const float* __restrict__ X, const float* __restrict__ val,
       const int* __restrict__ src, const int* __restrict__ dst,
       float* __restrict__ out, int nedges) {
  long gid = (long)blockIdx.x * blockDim.x + threadIdx.x;
  int e = (int)(gid >> 5);
  int lane = threadIdx.x & 31;
  if (e >= nedges) return;
  int es = __builtin_amdgcn_readfirstlane(e);  // provably wave-uniform
  int s = src[es];
  int d = dst[es];
  float v = val[es];
  float2 x = *((const float2*)(X + (size_t)s * DIM) + lane);
  float* op = out + (size_t)d * DIM + lane * 2;
  atomicAddF(op,     v * x.x);
  atomicAddF(op + 1, v * x.y);
}

// ---------------- column sums of concat(user_emb, item_emb) ----------------
__global__ void __launch_bounds__(256)
k_colsum(const float* __restrict__ u, int nu, const float* __restrict__ it, int ni,
         float* __restrict__ sums) {
  __shared__ float sh[256];
  int c = threadIdx.x & 63;
  int rgrp = threadIdx.x >> 6;  // 0..3
  int total = nu + ni;
  float s = 0.f;
  for (int r = blockIdx.x * 4 + rgrp; r < total; r += gridDim.x * 4) {
    const float* p = (r < nu) ? (u + (size_t)r * DIM) : (it + (size_t)(r - nu) * DIM);
    s += p[c];
  }
  sh[threadIdx.x] = s;
  __syncthreads();
  if (threadIdx.x < 64) {
    float t = sh[threadIdx.x] + sh[threadIdx.x + 64] + sh[threadIdx.x + 128] + sh[threadIdx.x + 192];
    atomicAddF(&sums[threadIdx.x], t);
  }
}

// ---------------- total sum of squares of (x - colmean) ----------------
__global__ void __launch_bounds__(256)
k_ss(const float* __restrict__ u, int nu, const float* __restrict__ it, int ni,
     const float* __restrict__ sums, float* __restrict__ ss, float inv_rows) {
  __shared__ float sh[256];
  int c = threadIdx.x & 63;
  int rgrp = threadIdx.x >> 6;
  float mean = sums[c] * inv_rows;
  int total = nu + ni;
  float s = 0.f;
  for (int r = blockIdx.x * 4 + rgrp; r < total; r += gridDim.x * 4) {
    const float* p = (r < nu) ? (u + (size_t)r * DIM) : (it + (size_t)(r - nu) * DIM);
    float d = p[c] - mean;
    s += d * d;
  }
  sh[threadIdx.x] = s;
  __syncthreads();
  for (int off = 128; off > 0; off >>= 1) {
    if (threadIdx.x < off) sh[threadIdx.x] += sh[threadIdx.x + off];
    __syncthreads();
  }
  if (threadIdx.x == 0) atomicAddF(ss, sh[0]);
}

// ---------------- normed = (x - mean) / sqrt(ss/rows + 1e-6), dual-dest ----------------
__global__ void __launch_bounds__(256)
k_normalize(const float* __restrict__ x, long n,
            const float* __restrict__ sums, const float* __restrict__ ss,
            float inv_rows, float* __restrict__ o1, float* __restrict__ o2) {
  long i = (long)blockIdx.x * blockDim.x + threadIdx.x;
  if (i >= n) return;
  int c = (int)(i & 63);
  float mean = sums[c] * inv_rows;
  float rn = sqrtf(ss[0] * inv_rows + 1e-6f);
  float v = (x[i] - mean) * (1.0f / rn);  // NORM_SCALE = 1.0
  o1[i] = v;
  if (o2) o2[i] = v;
}

// ---------------- acc += g ----------------
__global__ void __launch_bounds__(256)
k_add(float* __restrict__ acc, const float* __restrict__ g, long n) {
  long i = (long)blockIdx.x * blockDim.x + threadIdx.x;
  if (i < n) acc[i] += g[i];
}

// ---------------- out = acc/3 + 0.55*l2norm(fa) + 0.55*l2norm(fb), one wave per row ----------------
__global__ void __launch_bounds__(256)
k_final(const float* __restrict__ acc, const float* __restrict__ fa,
        const float* __restrict__ fb, float* __restrict__ out, int rows) {
  long gid = (long)blockIdx.x * blockDim.x + threadIdx.x;
  int w = (int)(gid >> 5);
  int lane = threadIdx.x & 31;
  if (w >= rows) return;
  size_t base = (size_t)w * DIM + lane * 2;
  float2 a = *(const float2*)(fa + base);
  float2 b = *(const float2*)(fb + base);
  float sa = a.x * a.x + a.y * a.y;
  float sb = b.x * b.x + b.y * b.y;
#pragma unroll
  for (int mlt = 16; mlt >= 1; mlt >>= 1) {
    sa += __shfl_xor(sa, mlt, 32);
    sb += __shfl_xor(sb, mlt, 32);
  }
  float ina = 1.0f / fmaxf(sqrtf(sa), 1e-12f);
  float inb = 1.0f / fmaxf(sqrtf(sb), 1e-12f);
  float2 ac = *(const float2*)(acc + base);
  float2 o;
  o.x = ac.x * (1.0f / 3.0f) + 0.55f * a.x * ina + 0.55f * b.x * inb;
  o.y = ac.y * (1.0f / 3.0f) + 0.55f * a.y * ina + 0.55f * b.y * inb;
  *(float2*)(out + base) = o;
}

static inline int cdiv(long a, long b) { return (int)((a + b - 1) / b); }

extern "C" void kernel_launch(void* const* d_in, const int* in_sizes, int n_in,
                              void* d_out, int out_size, void* d_ws, size_t ws_size,
                              hipStream_t stream) {
  (void)n_in; (void)out_size; (void)ws_size;

  const float* image_feats = (const float*)d_in[0];
  const float* text_feats  = (const float*)d_in[1];
  const float* user_emb    = (const float*)d_in[2];
  const float* item_emb    = (const float*)d_in[3];
  const float* W_img       = (const float*)d_in[4];
  const float* b_img       = (const float*)d_in[5];
  const float* W_txt       = (const float*)d_in[6];
  const float* b_txt       = (const float*)d_in[7];
  const float* val_ui      = (const float*)d_in[8];
  const float* val_iu      = (const float*)d_in[9];
  const int*   edge_u      = (const int*)d_in[10];
  const int*   edge_i      = (const int*)d_in[11];
  const int nedges = in_sizes[8];

  const long NU = (long)N_USERS * DIM;   // 6.4M floats
  const long NI = (long)N_ITEMS * DIM;   // 3.2M floats

  // d_out regions, return order: u_out, i_out, image_item, text_item, image_user, text_user
  float* out        = (float*)d_out;
  float* u_out      = out;
  float* i_out      = u_out + NU;
  float* image_item = i_out + NI;
  float* text_item  = image_item + NI;
  float* image_user = text_item + NI;
  float* text_user  = image_user + NU;

  // workspace layout
  float* ws = (float*)d_ws;
  float* feat_img = ws;               // NI
  float* feat_txt = feat_img + NI;    // NI
  float* u_t      = feat_txt + NI;    // NU
  float* iA       = u_t + NU;         // NI
  float* iB       = iA + NI;          // NI
  float* u_acc    = iB + NI;          // NU
  float* i_acc    = u_acc + NU;       // NI
  float* sums     = i_acc + NI;       // 64
  float* ssp      = sums + 64;        // 1

  const float inv_rows = 1.0f / (float)(N_USERS + N_ITEMS);
  dim3 B(256);
  const int gemm_grid = cdiv((long)N_ITEMS / 16, 8);        // 8 waves/block
  const int spmm_grid = cdiv((long)nedges * 32, 256);

  // ---- dense projections (WMMA) ----
  k_gemm_bias_wmma<<<gemm_grid, B, 0, stream>>>(image_feats, W_img, b_img, feat_img, N_ITEMS, IMG_D);
  k_gemm_bias_wmma<<<gemm_grid, B, 0, stream>>>(text_feats,  W_txt, b_txt, feat_txt, N_ITEMS, TXT_D);

  // ---- modality propagation: user then item ----
  k_zero<<<cdiv(NU, 256), B, 0, stream>>>(image_user, NU);
  k_spmm<<<spmm_grid, B, 0, stream>>>(feat_img, val_ui, edge_i, edge_u, image_user, nedges);
  k_zero<<<cdiv(NI, 256), B, 0, stream>>>(image_item, NI);
  k_spmm<<<spmm_grid, B, 0, stream>>>(image_user, val_iu, edge_u, edge_i, image_item, nedges);

  k_zero<<<cdiv(NU, 256), B, 0, stream>>>(text_user, NU);
  k_spmm<<<spmm_grid, B, 0, stream>>>(feat_txt, val_ui, edge_i, edge_u, text_user, nedges);
  k_zero<<<cdiv(NI, 256), B, 0, stream>>>(text_item, NI);
  k_spmm<<<spmm_grid, B, 0, stream>>>(text_user, val_iu, edge_u, edge_i, text_item, nedges);

  // ---- global mean / rownorm-mean normalization ----
  k_zero<<<1, B, 0, stream>>>(sums, 65);
  k_colsum<<<512, B, 0, stream>>>(user_emb, N_USERS, item_emb, N_ITEMS, sums);
  k_ss<<<512, B, 0, stream>>>(user_emb, N_USERS, item_emb, N_ITEMS, sums, ssp, inv_rows);
  k_normalize<<<cdiv(NU, 256), B, 0, stream>>>(user_emb, NU, sums, ssp, inv_rows, u_acc, nullptr);
  k_normalize<<<cdiv(NI, 256), B, 0, stream>>>(item_emb, NI, sums, ssp, inv_rows, iA, i_acc);

  // ---- GNN layer 1: u_t = spmm_ui(iA); iB = spmm_iu(u_t) ----
  k_zero<<<cdiv(NU, 256), B, 0, stream>>>(u_t, NU);
  k_spmm<<<spmm_grid, B, 0, stream>>>(iA, val_ui, edge_i, edge_u, u_t, nedges);
  k_add<<<cdiv(NU, 256), B, 0, stream>>>(u_acc, u_t, NU);
  k_zero<<<cdiv(NI, 256), B, 0, stream>>>(iB, NI);
  k_spmm<<<spmm_grid, B, 0, stream>>>(u_t, val_iu, edge_u, edge_i, iB, nedges);
  k_add<<<cdiv(NI, 256), B, 0, stream>>>(i_acc, iB, NI);

  // ---- GNN layer 2: u_t = spmm_ui(iB); iA = spmm_iu(u_t) ----
  k_zero<<<cdiv(NU, 256), B, 0, stream>>>(u_t, NU);
  k_spmm<<<spmm_grid, B, 0, stream>>>(iB, val_ui, edge_i, edge_u, u_t, nedges);
  k_add<<<cdiv(NU, 256), B, 0, stream>>>(u_acc, u_t, NU);
  k_zero<<<cdiv(NI, 256), B, 0, stream>>>(iA, NI);
  k_spmm<<<spmm_grid, B, 0, stream>>>(u_t, val_iu, edge_u, edge_i, iA, nedges);
  k_add<<<cdiv(NI, 256), B, 0, stream>>>(i_acc, iA, NI);

  // ---- final combine with l2-normed modality embeddings ----
  k_final<<<cdiv((long)N_USERS * 32, 256), B, 0, stream>>>(u_acc, image_user, text_user, u_out, N_USERS);
  k_final<<<cdiv((long)N_ITEMS * 32, 256), B, 0, stream>>>(i_acc, image_item, text_item, i_out, N_ITEMS);
}